// GraphTransformerNet_40235253628948
// MI455X (gfx1250) — compile-verified
//
#include <hip/hip_runtime.h>
#include <hip/hip_bf16.h>
#include <math.h>

// Problem constants (fixed by the reference)
#define NN 50000
#define EE 800000
#define GG 64
#define IND 128
#define HCD 192   // H*C = 3*64

typedef __attribute__((ext_vector_type(2))) float v2f;
typedef __attribute__((ext_vector_type(8))) float v8f;

// ---------------------------------------------------------------------------
// Utility: float atomic max via CAS (monotone, init to -inf)
// ---------------------------------------------------------------------------
__device__ __forceinline__ void atomicMaxF(float* addr, float val) {
    unsigned int* ua = (unsigned int*)addr;
    unsigned int old = *ua;
    while (__uint_as_float(old) < val) {
        unsigned int prev = atomicCAS(ua, old, __float_as_uint(val));
        if (prev == old) break;
        old = prev;
    }
}

__device__ __forceinline__ float waveReduceSum(float t) {
    #pragma unroll
    for (int m = 16; m > 0; m >>= 1) t += __shfl_xor(t, m, 32);
    return t;
}

// ---------------------------------------------------------------------------
// Fill kernel (init amax/-inf, denom/agg/pooled/cnt to 0)
// ---------------------------------------------------------------------------
__global__ void fill_kernel(float* __restrict__ p, long n, float v) {
    long i = (long)blockIdx.x * blockDim.x + threadIdx.x;
    if (i < n) p[i] = v;
}

// ---------------------------------------------------------------------------
// Fused 4-way projection GEMM:  Out[mat] = act @ W[mat] + b[mat]
//   A: [NN, Kdim] f32,  W: [Kdim, 192] f32,  Out: 4 x [NN, 192]
// One wave per 16x16 output tile using V_WMMA_F32_16X16X4_F32 (full fp32).
// Tile index: nt (12 N-tiles) | mat (4) | mt (3125 M-tiles) -> 150000 waves.
// A frag layout (16x4 f32): lanes 0-15 hold K={k,k+1}, lanes 16-31 K={k+2,k+3}
// B frag layout (4x16 f32): v0/v1 = B[k(+2hi)][n], B[k+1(+2hi)][n]
// ---------------------------------------------------------------------------
__global__ __launch_bounds__(256) void proj_gemm_kernel(
    const float* __restrict__ A, int Kdim,
    const float* __restrict__ Wq, const float* __restrict__ bq,
    const float* __restrict__ Wk, const float* __restrict__ bk,
    const float* __restrict__ Wv, const float* __restrict__ bv,
    const float* __restrict__ Ws, const float* __restrict__ bs,
    float* __restrict__ Out)
{
    int wave = (int)((blockIdx.x * blockDim.x + threadIdx.x) >> 5);
    int lane = threadIdx.x & 31;
    int nt  = wave % 12;
    int mat = (wave / 12) & 3;
    int mt  = wave / 48;

    const float* W; const float* bvec;
    if      (mat == 0) { W = Wq; bvec = bq; }
    else if (mat == 1) { W = Wk; bvec = bk; }
    else if (mat == 2) { W = Wv; bvec = bv; }
    else               { W = Ws; bvec = bs; }
    float* O = Out + (size_t)mat * (size_t)NN * HCD;

    int laneLo = lane & 15;
    int hi     = (lane >> 4) & 1;
    int row = mt * 16 + laneLo;
    int col = nt * 16 + laneLo;

    const float* aPtr = A + (size_t)row * Kdim + 2 * hi;
    const float* wPtr = W + (size_t)(2 * hi) * HCD + col;

    v8f acc = {};
    for (int k = 0; k < Kdim; k += 4) {
        v2f a, b;
        a.x = aPtr[k];
        a.y = aPtr[k + 1];
        b.x = wPtr[(size_t)k * HCD];
        b.y = wPtr[(size_t)(k + 1) * HCD];
        acc = __builtin_amdgcn_wmma_f32_16x16x4_f32(
            false, a, false, b, (short)0, acc, false, false);
    }

    float bb = bvec[col];
    int rbase = mt * 16 + 8 * hi;
    #pragma unroll
    for (int i = 0; i < 8; ++i)
        O[(size_t)(rbase + i) * HCD + col] = acc[i] + bb;
}

// ---------------------------------------------------------------------------
// Edge pass A: alpha[e][h] = (Q[dst] . K[src])_h / sqrt(64); segment max
// One wave per edge; each lane covers 2 elems per head (64 ch / 32 lanes).
// ---------------------------------------------------------------------------
__global__ __launch_bounds__(256) void edge_alpha_kernel(
    const float* __restrict__ Q, const float* __restrict__ Kp,
    const int* __restrict__ src, const int* __restrict__ dst,
    float* __restrict__ alpha, float* __restrict__ amax)
{
    int e = (int)((blockIdx.x * blockDim.x + threadIdx.x) >> 5);
    int lane = threadIdx.x & 31;
    if (e >= EE) return;
    int s = src[e], d = dst[e];
    const float* q = Q  + (size_t)d * HCD;
    const float* k = Kp + (size_t)s * HCD;
    float p[3];
    #pragma unroll
    for (int h = 0; h < 3; ++h) {
        int i0 = h * 64 + lane;
        float t = q[i0] * k[i0] + q[i0 + 32] * k[i0 + 32];
        p[h] = waveReduceSum(t) * 0.125f;   // 1/sqrt(C), C=64
    }
    if (lane < 3) {
        alpha[(size_t)e * 3 + lane] = p[lane];
        atomicMaxF(&amax[(size_t)d * 3 + lane], p[lane]);
    }
}

// ---------------------------------------------------------------------------
// Edge pass B: ex = exp(alpha - amax'); denom[d] += ex; agg[d] += ex * V[src]
// (normalization folded into node epilogue: identical to a = ex/denom math)
// ---------------------------------------------------------------------------
__global__ __launch_bounds__(256) void edge_scatter_kernel(
    const float* __restrict__ V,
    const int* __restrict__ src, const int* __restrict__ dst,
    const float* __restrict__ alpha, const float* __restrict__ amax,
    float* __restrict__ denom, float* __restrict__ agg)
{
    int e = (int)((blockIdx.x * blockDim.x + threadIdx.x) >> 5);
    int lane = threadIdx.x & 31;
    if (e >= EE) return;
    int s = src[e], d = dst[e];
    float ex[3];
    #pragma unroll
    for (int h = 0; h < 3; ++h) {
        float am = amax[(size_t)d * 3 + h];
        if (!(am > -1e37f)) am = 0.0f;             // where(isfinite, amax, 0)
        ex[h] = expf(alpha[(size_t)e * 3 + h] - am);
    }
    if (lane < 3) atomicAdd(&denom[(size_t)d * 3 + lane], ex[lane]);
    const float* v = V   + (size_t)s * HCD;
    float*       o = agg + (size_t)d * HCD;
    #pragma unroll
    for (int j = 0; j < 6; ++j) {
        int idx = lane + 32 * j;                   // head = j>>1
        atomicAdd(&o[idx], ex[j >> 1] * v[idx]);
    }
}

// ---------------------------------------------------------------------------
// Node epilogue: attn = agg/(denom+1e-16); beta-gated skip with R; ELU.
// Writes activation back into agg (in place). One wave per node.
// ---------------------------------------------------------------------------
__global__ __launch_bounds__(256) void node_update_kernel(
    const float* __restrict__ R, const float* __restrict__ Wb,
    const float* __restrict__ denom, float* __restrict__ agg)
{
    int n = (int)((blockIdx.x * blockDim.x + threadIdx.x) >> 5);
    int lane = threadIdx.x & 31;
    if (n >= NN) return;
    float dn[3];
    #pragma unroll
    for (int h = 0; h < 3; ++h) dn[h] = denom[(size_t)n * 3 + h] + 1e-16f;

    float attn[6], r[6];
    float g = 0.0f;
    #pragma unroll
    for (int j = 0; j < 6; ++j) {
        int idx = lane + 32 * j;
        float a  = agg[(size_t)n * HCD + idx] / dn[j >> 1];
        float rr = R[(size_t)n * HCD + idx];
        attn[j] = a; r[j] = rr;
        g += Wb[idx] * a + Wb[192 + idx] * rr + Wb[384 + idx] * (a - rr);
    }
    g = waveReduceSum(g);
    float beta = 1.0f / (1.0f + expf(-g));
    #pragma unroll
    for (int j = 0; j < 6; ++j) {
        int idx = lane + 32 * j;
        float h = beta * r[j] + (1.0f - beta) * attn[j];
        h = (h > 0.0f) ? h : (expf(h) - 1.0f);     // ELU(alpha=1)
        agg[(size_t)n * HCD + idx] = h;
    }
}

// ---------------------------------------------------------------------------
// Pool: pooled[batch[n]] += h[n]; cnt[batch[n]] += 1
// ---------------------------------------------------------------------------
__global__ __launch_bounds__(256) void pool_kernel(
    const float* __restrict__ h, const int* __restrict__ batch,
    float* __restrict__ pooled, float* __restrict__ cnt)
{
    int n = (int)((blockIdx.x * blockDim.x + threadIdx.x) >> 5);
    int lane = threadIdx.x & 31;
    if (n >= NN) return;
    int g = batch[n];
    #pragma unroll
    for (int j = 0; j < 6; ++j) {
        int idx = lane + 32 * j;
        atomicAdd(&pooled[(size_t)g * HCD + idx], h[(size_t)n * HCD + idx]);
    }
    if (lane == 0) atomicAdd(&cnt[g], 1.0f);
}

// ---------------------------------------------------------------------------
// Final: logits[g] = (pooled[g] . Wlin) / max(cnt,1) + blin
// ---------------------------------------------------------------------------
__global__ __launch_bounds__(256) void final_kernel(
    const float* __restrict__ pooled, const float* __restrict__ cnt,
    const float* __restrict__ Wlin, const float* __restrict__ blin,
    float* __restrict__ out)
{
    int g = (int)((blockIdx.x * blockDim.x + threadIdx.x) >> 5);
    int lane = threadIdx.x & 31;
    if (g >= GG) return;
    float t = 0.0f;
    #pragma unroll
    for (int j = 0; j < 6; ++j) {
        int idx = lane + 32 * j;
        t += pooled[(size_t)g * HCD + idx] * Wlin[idx];
    }
    t = waveReduceSum(t);
    if (lane == 0) {
        float c = cnt[g]; c = fmaxf(c, 1.0f);
        out[g] = t / c + blin[0];
    }
}

// ---------------------------------------------------------------------------
// Host launcher
// ---------------------------------------------------------------------------
extern "C" void kernel_launch(void* const* d_in, const int* in_sizes, int n_in,
                              void* d_out, int out_size, void* d_ws, size_t ws_size,
                              hipStream_t stream) {
    (void)in_sizes; (void)n_in; (void)out_size; (void)ws_size;
    const float* x     = (const float*)d_in[0];
    const int*   ei    = (const int*)d_in[1];
    const int*   batch = (const int*)d_in[2];
    const int* src = ei;
    const int* dst = ei + EE;

    const float* Wmat[2][4]; const float* bvec[2][4]; const float* Wb[2];
    for (int l = 0; l < 2; ++l) {
        int base = 3 + l * 9;
        for (int m = 0; m < 4; ++m) {
            Wmat[l][m] = (const float*)d_in[base + 2 * m];
            bvec[l][m] = (const float*)d_in[base + 2 * m + 1];
        }
        Wb[l] = (const float*)d_in[base + 8];
    }
    const float* Wlin = (const float*)d_in[21];
    const float* blin = (const float*)d_in[22];

    // Workspace layout (floats)
    const size_t NP = (size_t)NN * HCD;          // 9.6M
    float* ws     = (float*)d_ws;
    float* proj   = ws;                           // 4*NP  (Q,K,V,R)
    float* alpha  = proj + 4 * NP;                // EE*3
    float* amax   = alpha + (size_t)EE * 3;       // NN*3
    float* denom  = amax + (size_t)NN * 3;        // NN*3
    float* agg    = denom + (size_t)NN * 3;       // NP  (also activation h)
    float* pooled = agg + NP;                     // GG*HCD
    float* cnt    = pooled + (size_t)GG * HCD;    // GG

    float* Qp = proj;
    float* Kp = proj + NP;
    float* Vp = proj + 2 * NP;
    float* Rp = proj + 3 * NP;

    const int gemmBlocks = ((NN / 16) * 48) / 8;  // 18750 blocks, 8 waves each
    const int edgeBlocks = EE / 8;                // 100000
    const int nodeBlocks = NN / 8;                // 6250

    for (int l = 0; l < 2; ++l) {
        const float* act = (l == 0) ? x : agg;
        int Kdim = (l == 0) ? IND : HCD;

        proj_gemm_kernel<<<gemmBlocks, 256, 0, stream>>>(
            act, Kdim,
            Wmat[l][0], bvec[l][0], Wmat[l][1], bvec[l][1],
            Wmat[l][2], bvec[l][2], Wmat[l][3], bvec[l][3], proj);

        // init segment buffers AFTER gemm (layer 1 reads agg as activation)
        fill_kernel<<<((long)NN * 3 + 255) / 256, 256, 0, stream>>>(
            amax, (long)NN * 3, -__builtin_inff());
        fill_kernel<<<((long)(NN * 3 + NP) + 255) / 256, 256, 0, stream>>>(
            denom, (long)(NN * 3 + NP), 0.0f);   // denom + agg contiguous

        edge_alpha_kernel<<<edgeBlocks, 256, 0, stream>>>(
            Qp, Kp, src, dst, alpha, amax);
        edge_scatter_kernel<<<edgeBlocks, 256, 0, stream>>>(
            Vp, src, dst, alpha, amax, denom, agg);
        node_update_kernel<<<nodeBlocks, 256, 0, stream>>>(
            Rp, Wb[l], denom, agg);
    }

    fill_kernel<<<((long)(GG * HCD + GG) + 255) / 256, 256, 0, stream>>>(
        pooled, (long)(GG * HCD + GG), 0.0f);    // pooled + cnt contiguous
    pool_kernel<<<nodeBlocks, 256, 0, stream>>>(agg, batch, pooled, cnt);
    final_kernel<<<8, 256, 0, stream>>>(pooled, cnt, Wlin, blin, (float*)d_out);
}